// StreamingDWRBlock_80968723464195
// MI455X (gfx1250) — compile-verified
//
#include <hip/hip_runtime.h>
#include <cmath>

// ---------------- problem constants (match reference) ----------------
#define T_TOK   2048      // B*S tokens
#define DMODEL  1024      // D
#define NEXP    8         // E experts
#define FFN     2048      // F hidden
#define ROWS_CAP 4608     // sum_e padTo64(count_e) <= 4096 + 8*63 = 4600

typedef __attribute__((ext_vector_type(2))) float v2f;
typedef __attribute__((ext_vector_type(8))) float v8f;

#define AS_STRIDE 36      // A LDS row stride (dwords): 16B-aligned rows
#define BS_STRIDE 144     // B LDS row stride (dwords): 16B aligned, 2*144%64=32
                          // -> lane halves (K, K+2) hit disjoint bank sets

__device__ __forceinline__ float gelu_tanh(float v) {
    // jax.nn.gelu default (approximate=True)
    const float c = 0.7978845608028654f;
    float u = c * (v + 0.044715f * v * v * v);
    return 0.5f * v * (1.0f + tanhf(u));
}

// ---------------------------------------------------------------------
// Kernel 1: router. One wave (32 lanes) per token: logits = x@Wr + br,
// softmax over E=8, top-2 (ties -> lower index, like jax.lax.top_k).
// ---------------------------------------------------------------------
__global__ __launch_bounds__(256) void router_kernel(
    const float* __restrict__ x, const float* __restrict__ Wr,
    const float* __restrict__ br, float* __restrict__ gates,
    int* __restrict__ eidx)
{
    int wave = blockIdx.x * (blockDim.x >> 5) + (threadIdx.x >> 5);
    int lane = threadIdx.x & 31;
    if (wave >= T_TOK) return;  // wave-uniform

    float acc[NEXP];
#pragma unroll
    for (int e = 0; e < NEXP; ++e) acc[e] = 0.0f;

    const float* xrow = x + (size_t)wave * DMODEL;
#pragma unroll 4
    for (int it = 0; it < DMODEL / 32; ++it) {
        int d = it * 32 + lane;
        float xv = xrow[d];
        const float* wr = Wr + (size_t)d * NEXP;
        float4 w0 = *(const float4*)(wr);
        float4 w1 = *(const float4*)(wr + 4);
        acc[0] += xv * w0.x; acc[1] += xv * w0.y;
        acc[2] += xv * w0.z; acc[3] += xv * w0.w;
        acc[4] += xv * w1.x; acc[5] += xv * w1.y;
        acc[6] += xv * w1.z; acc[7] += xv * w1.w;
    }
#pragma unroll
    for (int e = 0; e < NEXP; ++e)
        for (int off = 16; off >= 1; off >>= 1)
            acc[e] += __shfl_xor(acc[e], off, 32);

    if (lane == 0) {
        float l[NEXP];
#pragma unroll
        for (int e = 0; e < NEXP; ++e) l[e] = acc[e] + br[e];
        int i1 = 0;
#pragma unroll
        for (int e = 1; e < NEXP; ++e) if (l[e] > l[i1]) i1 = e;
        int i2 = (i1 == 0) ? 1 : 0;
#pragma unroll
        for (int e = 0; e < NEXP; ++e)
            if (e != i1 && l[e] > l[i2]) i2 = e;
        float mx = l[i1];
        float s = 0.0f;
#pragma unroll
        for (int e = 0; e < NEXP; ++e) s += expf(l[e] - mx);
        gates[2 * wave + 0] = expf(l[i1] - mx) / s;
        gates[2 * wave + 1] = expf(l[i2] - mx) / s;
        eidx[2 * wave + 0] = i1;
        eidx[2 * wave + 1] = i2;
    }
}

// ---------------------------------------------------------------------
// Kernel 2: deterministic compaction (single block). Builds per-expert
// row lists padded to 64 rows, inverse map token_of_row (-1 -> pad row),
// and pair_row[t][k] so the final kernel can gather (no atomics).
// meta: [0..7]=count [8..15]=padcount [16..23]=rowbase
// ---------------------------------------------------------------------
__global__ __launch_bounds__(256) void compact_kernel(
    const int* __restrict__ eidx, int* __restrict__ pair_row,
    int* __restrict__ token_of_row, int* __restrict__ meta)
{
    __shared__ int s_cnt[NEXP];
    __shared__ int s_base[NEXP];
    int tid = threadIdx.x;

    for (int r = tid; r < ROWS_CAP; r += blockDim.x) token_of_row[r] = -1;
    __syncthreads();

    if (tid < NEXP) {
        int c = 0;
        for (int i = 0; i < 2 * T_TOK; ++i) if (eidx[i] == tid) c++;
        s_cnt[tid] = c;
    }
    __syncthreads();

    if (tid == 0) {
        int base = 0;
        for (int e = 0; e < NEXP; ++e) {
            int pc = (s_cnt[e] + 63) & ~63;
            meta[e] = s_cnt[e];
            meta[8 + e] = pc;
            meta[16 + e] = base;
            s_base[e] = base;
            base += pc;
        }
    }
    __syncthreads();

    if (tid < NEXP) {
        int pos = s_base[tid];
        for (int i = 0; i < 2 * T_TOK; ++i) {
            if (eidx[i] == tid) {
                pair_row[i] = pos;
                token_of_row[pos] = i >> 1;
                pos++;
            }
        }
    }
}

// ---------------------------------------------------------------------
// Kernel 3/4: fp32 WMMA GEMM on gathered rows, double-buffered LDS with
// GLOBAL_LOAD_ASYNC_TO_LDS_B128 staging (ASYNCcnt) for both A and B.
//   MODE 1: A = x rows via token_of_row (pad rows read token 0; their
//           outputs are finite and never gathered), C = gelu(A@W1+b1) -> h
//   MODE 2: A = h rows (dense), C = A@W2[e] + b2[e] -> y
// Block 256 thr = 8 waves; tile BM=64 x BN=128, BK=32. Wave (w>>1,w&1)
// owns 16x64 -> 4 v8f accs; 32 v_wmma_f32_16x16x4_f32 per K-chunk.
// ---------------------------------------------------------------------
template <int MODE>
__global__ __launch_bounds__(256) void moe_gemm_kernel(
    const float* __restrict__ Abase,   // MODE1: x (T,D); MODE2: h (ROWS_CAP,F)
    const float* __restrict__ Bbase,   // W1 (E,D,F) or W2 (E,F,D)
    const float* __restrict__ biasBase,// b1 (E,F) or b2 (E,D)
    float* __restrict__ Out,           // h (ROWS_CAP,F) or y (ROWS_CAP,D)
    const int* __restrict__ meta,
    const int* __restrict__ token_of_row,
    int K, int N)
{
    const int NT = N >> 7;        // N/128
    const int MT = T_TOK / 64;
    int bx = blockIdx.x;
    int e = bx / (MT * NT);
    int rem = bx - e * (MT * NT);
    int mt = rem / NT, nt = rem - mt * NT;

    int padc = meta[8 + e];
    int m0 = mt * 64;
    if (m0 >= padc) return;                    // block-uniform: EXEC stays full
    int rowbase = meta[16 + e];
    int n0 = nt * 128;

    const float* Bg   = Bbase    + (size_t)e * K * N;
    const float* bias = biasBase + (size_t)e * N;

    __shared__ float As[2][64 * AS_STRIDE];
    __shared__ float Bs[2][32 * BS_STRIDE];

    int tid = threadIdx.x;
    int lane = tid & 31, wave = tid >> 5;
    int lm = lane & 15, lh = lane >> 4;
    int mw = (wave >> 1) * 16;                 // 0,16,32,48
    int nw = (wave & 1) * 64;                  // 0,64

    // Per-thread A staging row is fixed across the K loop.
    int am0 = tid >> 2;                        // 0..63
    int ak  = (tid & 3) * 8;                   // 0,8,16,24
    const float* aptr;
    if (MODE == 1) {
        int tok = token_of_row[rowbase + m0 + am0];
        if (tok < 0) tok = 0;                  // pad rows: finite, never read
        aptr = Abase + (size_t)tok * K;
    } else {
        aptr = Abase + (size_t)(rowbase + m0 + am0) * K;
    }
    int bnn = (tid & 31) * 4;                  // B staging column

    uint32_t asB[2], bsB[2];
    asB[0] = (uint32_t)(uintptr_t)&As[0][0];
    asB[1] = (uint32_t)(uintptr_t)&As[1][0];
    bsB[0] = (uint32_t)(uintptr_t)&Bs[0][0];
    bsB[1] = (uint32_t)(uintptr_t)&Bs[1][0];

    // Async stage of one 64x32 A tile + 32x128 B tile into buffer `buf`.
    auto issue_stage = [&](int k0, int buf) {
        uint32_t lA = asB[buf] + (uint32_t)((am0 * AS_STRIDE + ak) * 4);
        unsigned long long gA = (unsigned long long)(uintptr_t)(aptr + k0 + ak);
        asm volatile("global_load_async_to_lds_b128 %0, %1, off"
                     :: "v"(lA), "v"(gA) : "memory");
        asm volatile("global_load_async_to_lds_b128 %0, %1, off offset:16"
                     :: "v"(lA), "v"(gA) : "memory");
#pragma unroll
        for (int p = 0; p < 4; ++p) {
            int kk = p * 8 + wave;
            uint32_t lB = bsB[buf] + (uint32_t)((kk * BS_STRIDE + bnn) * 4);
            unsigned long long gB = (unsigned long long)(uintptr_t)
                (Bg + (size_t)(k0 + kk) * N + n0 + bnn);
            asm volatile("global_load_async_to_lds_b128 %0, %1, off"
                         :: "v"(lB), "v"(gB) : "memory");
        }
    };

    v8f acc[4];
#pragma unroll
    for (int j = 0; j < 4; ++j)
#pragma unroll
        for (int i = 0; i < 8; ++i) acc[j][i] = 0.0f;

    issue_stage(0, 0);
    asm volatile("s_wait_asynccnt 0" ::: "memory");
    __syncthreads();

    const int KT = K / 32;
    for (int kt = 0; kt < KT; ++kt) {
        int cur = kt & 1;
        if (kt + 1 < KT) issue_stage((kt + 1) * 32, cur ^ 1);

        const float* AsCur = &As[cur][0];
        const float* BsCur = &Bs[cur][0];
#pragma unroll
        for (int ks = 0; ks < 8; ++ks) {
            // A frag: lane l, reg r -> A[M=l%16][K=2*(l/16)+r]
            v2f a = *(const v2f*)&AsCur[(mw + lm) * AS_STRIDE + ks * 4 + 2 * lh];
            int kr = ks * 4 + 2 * lh;
#pragma unroll
            for (int j = 0; j < 4; ++j) {
                int c = nw + 16 * j + lm;
                v2f b;
                b[0] = BsCur[kr * BS_STRIDE + c];                // K = kr
                b[1] = BsCur[kr * BS_STRIDE + BS_STRIDE + c];    // K = kr+1
                acc[j] = __builtin_amdgcn_wmma_f32_16x16x4_f32(
                    false, a, false, b, (short)0, acc[j], false, false);
            }
        }
        asm volatile("s_wait_asynccnt 0" ::: "memory");
        __syncthreads();
    }

    // --- epilogue: bias (+gelu for MODE1), store ---
#pragma unroll
    for (int j = 0; j < 4; ++j) {
        int col = n0 + nw + 16 * j + lm;
        float bs = bias[col];
#pragma unroll
        for (int r = 0; r < 8; ++r) {
            int mrow = mw + 8 * lh + r;          // C/D: M = r + 8*(l/16)
            size_t orow = (size_t)(rowbase + m0 + mrow);
            float v = acc[j][r] + bs;
            if (MODE == 1) v = gelu_tanh(v);
            Out[orow * (size_t)N + col] = v;
        }
    }
}

// ---------------------------------------------------------------------
// Kernel 5: gather-gated combine + residual + LayerNorm. One block per
// token; each token has exactly two pair rows (deterministic gather).
// ---------------------------------------------------------------------
__global__ __launch_bounds__(256) void combine_ln_kernel(
    const float* __restrict__ x, const float* __restrict__ y,
    const float* __restrict__ gates, const int* __restrict__ pair_row,
    const float* __restrict__ gamma, const float* __restrict__ beta,
    float* __restrict__ out)
{
    __shared__ float red[256];
    int t = blockIdx.x;
    int tid = threadIdx.x;

    int r0 = pair_row[2 * t + 0];
    int r1 = pair_row[2 * t + 1];
    float g0 = gates[2 * t + 0];
    float g1 = gates[2 * t + 1];

    const float* xr = x + (size_t)t * DMODEL;
    const float* y0 = y + (size_t)r0 * DMODEL;
    const float* y1 = y + (size_t)r1 * DMODEL;

    float vals[4];
    float lsum = 0.0f;
#pragma unroll
    for (int i = 0; i < 4; ++i) {
        int d = tid + 256 * i;
        float v = xr[d] + g0 * y0[d] + g1 * y1[d];
        vals[i] = v;
        lsum += v;
    }
    red[tid] = lsum;
    __syncthreads();
    for (int s = 128; s > 0; s >>= 1) {
        if (tid < s) red[tid] += red[tid + s];
        __syncthreads();
    }
    float mu = red[0] * (1.0f / DMODEL);
    __syncthreads();

    float lvar = 0.0f;
#pragma unroll
    for (int i = 0; i < 4; ++i) {
        float df = vals[i] - mu;
        lvar += df * df;
    }
    red[tid] = lvar;
    __syncthreads();
    for (int s = 128; s > 0; s >>= 1) {
        if (tid < s) red[tid] += red[tid + s];
        __syncthreads();
    }
    float rstd = rsqrtf(red[0] * (1.0f / DMODEL) + 1e-5f);

#pragma unroll
    for (int i = 0; i < 4; ++i) {
        int d = tid + 256 * i;
        out[(size_t)t * DMODEL + d] = (vals[i] - mu) * rstd * gamma[d] + beta[d];
    }
}

// ---------------------------------------------------------------------
extern "C" void kernel_launch(void* const* d_in, const int* in_sizes, int n_in,
                              void* d_out, int out_size, void* d_ws, size_t ws_size,
                              hipStream_t stream)
{
    const float* x     = (const float*)d_in[0];
    const float* Wr    = (const float*)d_in[1];
    const float* br    = (const float*)d_in[2];
    const float* W1    = (const float*)d_in[3];
    const float* b1    = (const float*)d_in[4];
    const float* W2    = (const float*)d_in[5];
    const float* b2    = (const float*)d_in[6];
    const float* gamma = (const float*)d_in[7];
    const float* beta  = (const float*)d_in[8];
    float* out = (float*)d_out;

    // Workspace carve-up (~56.8 MB total), 256B aligned regions.
    char* ws = (char*)d_ws;
    size_t off = 0;
    auto carve = [&](size_t bytes) {
        void* p = ws + off;
        off = (off + bytes + 255) & ~(size_t)255;
        return p;
    };
    float* gates        = (float*)carve(2 * T_TOK * sizeof(float));
    int*   eidx         = (int*)  carve(2 * T_TOK * sizeof(int));
    int*   pair_row     = (int*)  carve(2 * T_TOK * sizeof(int));
    int*   meta         = (int*)  carve(24 * sizeof(int));
    int*   token_of_row = (int*)  carve(ROWS_CAP * sizeof(int));
    float* h            = (float*)carve((size_t)ROWS_CAP * FFN * sizeof(float));
    float* y            = (float*)carve((size_t)ROWS_CAP * DMODEL * sizeof(float));
    (void)ws_size; (void)in_sizes; (void)n_in; (void)out_size;

    // 1) router: 2048 waves = 256 blocks x 8 waves
    router_kernel<<<T_TOK / 8, 256, 0, stream>>>(x, Wr, br, gates, eidx);

    // 2) deterministic compaction
    compact_kernel<<<1, 256, 0, stream>>>(eidx, pair_row, token_of_row, meta);

    // 3) GEMM1: h = gelu(x_gathered @ W1[e] + b1[e]); K=D, N=F
    {
        int blocks = NEXP * (T_TOK / 64) * (FFN / 128); // 4096 worst-case
        moe_gemm_kernel<1><<<blocks, 256, 0, stream>>>(
            x, W1, b1, h, meta, token_of_row, DMODEL, FFN);
    }
    // 4) GEMM2: y = h @ W2[e] + b2[e]; K=F, N=D
    {
        int blocks = NEXP * (T_TOK / 64) * (DMODEL / 128); // 2048 worst-case
        moe_gemm_kernel<2><<<blocks, 256, 0, stream>>>(
            h, W2, b2, y, meta, token_of_row, FFN, DMODEL);
    }
    // 5) gated combine + residual + LayerNorm
    combine_ln_kernel<<<T_TOK, 256, 0, stream>>>(
        x, y, gates, pair_row, gamma, beta, out);
}